// ModelNew_1580547974754
// MI455X (gfx1250) — compile-verified
//
#include <hip/hip_runtime.h>
#include <hip/hip_bf16.h>

// CDNA5 / gfx1250. Fused ConvTranspose2d(64->64,k4,s2,p1) + BN(inference) +
// channel softmax + 2x2 maxpool, built on V_WMMA_F32_16X16X4_F32.
//
// Decomposition: output parity phase (py,px) at plane coord (y,x) is a dense
// 2x2-tap conv:  iy = y + py - ty, kh = 2*ty + (1-py)   (same for x/kw).
// The 2x2 maxpool == max over the 4 phase planes at identical (y,x).
// One workgroup (4 wave32) = 4 phases over the same 16 pooled pixels.
// Per wave: GEMM D[64co x 16px] , K = 64ci * (4 taps) -> 64 K-steps x 4 M-tiles
// of v_wmma_f32_16x16x4_f32 (full f32 precision, matches f32 reference).
//
// K-pair ordering note: within a lane's K pair we store (tx=1, tx=0) so the
// B fragment is the ascending-address pair {xbuf[bcol], xbuf[bcol+1]} and can
// be consumed straight out of ds_load_2addr_b32 with no register swaps.

typedef __attribute__((ext_vector_type(2))) float v2f;
typedef __attribute__((ext_vector_type(8))) float v8f;

#define CHUNK 16   // input channels staged in LDS per iteration

// ---------------------------------------------------------------------------
// Prep: relayout weights (Cin,Cout,4,4) into per-phase WMMA A-fragments:
//   w2[ph][ci][tile][lane][2]  (v2f per lane, directly loadable as A operand)
// A-matrix 16x4 layout (ISA 7.12.2): lanes 0-15 hold K={0,1}, lanes 16-31 hold
// K={2,3}; M = lane&15.  K = 2*ty + {0: tx=1, 1: tx=0}  (pair order swapped to
// match the ascending-address B load; sum over K makes order irrelevant as
// long as A and B agree).
// ---------------------------------------------------------------------------
__global__ void relayout_weights(const float* __restrict__ w,
                                 float* __restrict__ w2) {
  int idx = blockIdx.x * blockDim.x + threadIdx.x;  // [ph][ci][tile][lane]
  if (idx >= 4 * 64 * 4 * 32) return;
  int lane = idx & 31;
  int tile = (idx >> 5) & 3;
  int ci   = (idx >> 7) & 63;
  int ph   = idx >> 13;
  int py = ph >> 1, px = ph & 1;
  int co = tile * 16 + (lane & 15);
  int ty = lane >> 4;
  int kh  = (ty ? 2 : 0) + (1 - py);
  int kw0 = 1 - px;       // tx = 0  (ix = xp + px)
  int kw1 = 3 - px;       // tx = 1  (ix = xp + px - 1)
  const float* wb = w + ((ci * 64 + co) * 4 + kh) * 4;  // (in,out,kh,kw)
  w2[idx * 2 + 0] = wb[kw1];   // pair slot 0 <- tx=1 (lower ix address)
  w2[idx * 2 + 1] = wb[kw0];   // pair slot 1 <- tx=0 (higher ix address)
}

// ---------------------------------------------------------------------------
// Main fused kernel. Grid: 32(n) * 64(y) * 4(xseg) blocks of 128 threads.
// ---------------------------------------------------------------------------
__global__ __launch_bounds__(128) void fused_tconv_bn_softmax_pool(
    const float* __restrict__ x, const float* __restrict__ w2,
    const float* __restrict__ cb, const float* __restrict__ gmm,
    const float* __restrict__ bta, const float* __restrict__ rmean,
    const float* __restrict__ rvar, float* __restrict__ out) {
  __shared__ float xbuf[CHUNK][3][20];   // rows y-1..y+1, cols x0-1..x0+16 (pad)
  __shared__ float red[4][64][16];       // per-phase softmax tiles for pool-max
  __shared__ float pA[64], pB[64];       // fused BN scale / bias

  const int bid  = blockIdx.x;
  const int xseg = bid & 3;
  const int y    = (bid >> 2) & 63;
  const int n    = bid >> 8;
  const int x0   = xseg * 16;

  const int t    = threadIdx.x;
  const int wave = t >> 5;               // = phase id
  const int lane = t & 31;
  const int py = wave >> 1, px = wave & 1;
  const int p    = lane & 15;            // pixel (N) index
  const int ty   = lane >> 4;            // which K-pair half this lane holds
  const int rrow = py - ty + 1;          // xbuf row: iy = y - 1 + rrow
  const int bcol = p + px;               // xbuf col of pair slot 0 (tx=1)

  if (t < 64) {
    float s = gmm[t] * rsqrtf(rvar[t] + 1e-5f);
    pA[t] = s;
    pB[t] = (cb[t] - rmean[t]) * s + bta[t];
  }

  v8f acc0 = {0.f, 0.f, 0.f, 0.f, 0.f, 0.f, 0.f, 0.f};
  v8f acc1 = acc0, acc2 = acc0, acc3 = acc0;

  // per-(wave,lane) base into relaid weights: index = ((ph*64+ci)*4+tile)*32+lane
  const v2f* wfrag = ((const v2f*)w2) + (wave << 13) + lane;
  const float* xn = x + (size_t)n * 64 * 64 * 64;

  for (int c0 = 0; c0 < 64; c0 += CHUNK) {
    // Stage CHUNK input channels: 3 rows x 18 cols, zero-padded borders.
    for (int i = t; i < CHUNK * 3 * 18; i += 128) {
      int cc  = i / 54;
      int rem = i - cc * 54;
      int rr  = rem / 18;
      int col = rem - rr * 18;
      int iy  = y - 1 + rr;
      int ix  = x0 - 1 + col;
      float v = 0.f;
      if (iy >= 0 && iy < 64 && ix >= 0 && ix < 64)
        v = xn[((c0 + cc) * 64 + iy) * 64 + ix];
      xbuf[cc][rr][col] = v;
    }
    __syncthreads();

#pragma unroll 4
    for (int cc = 0; cc < CHUNK; ++cc) {
      // B-fragment 4x16 (mirrors A layout): ascending-address pair, no swap.
      v2f b;
      b.x = xbuf[cc][rrow][bcol];       // pair slot 0: tx=1, ix = xp + px - 1
      b.y = xbuf[cc][rrow][bcol + 1];   // pair slot 1: tx=0, ix = xp + px
      const v2f* wp = wfrag + ((c0 + cc) << 7);   // + ci*4*32
      acc0 = __builtin_amdgcn_wmma_f32_16x16x4_f32(false, wp[0],  false, b,
                                                   (short)0, acc0, false, false);
      acc1 = __builtin_amdgcn_wmma_f32_16x16x4_f32(false, wp[32], false, b,
                                                   (short)0, acc1, false, false);
      acc2 = __builtin_amdgcn_wmma_f32_16x16x4_f32(false, wp[64], false, b,
                                                   (short)0, acc2, false, false);
      acc3 = __builtin_amdgcn_wmma_f32_16x16x4_f32(false, wp[96], false, b,
                                                   (short)0, acc3, false, false);
    }
    __syncthreads();
  }

  // Epilogue: BN affine + softmax over 64 channels per pixel.
  // C/D layout: VGPR r -> M = r (lanes 0-15) / r+8 (lanes 16-31); N = lane&15.
  // Lanes L and L+16 hold the two channel halves of the SAME pixel.
  float vals[32];
  float m = -3.4e38f;
#pragma unroll
  for (int tile = 0; tile < 4; ++tile) {
    v8f a = (tile == 0) ? acc0 : (tile == 1) ? acc1 : (tile == 2) ? acc2 : acc3;
#pragma unroll
    for (int r = 0; r < 8; ++r) {
      int c = tile * 16 + r + (ty ? 8 : 0);
      float v = a[r] * pA[c] + pB[c];
      vals[tile * 8 + r] = v;
      m = fmaxf(m, v);
    }
  }
  m = fmaxf(m, __shfl_xor(m, 16, 32));
  float s = 0.f;
#pragma unroll
  for (int i = 0; i < 32; ++i) {
    vals[i] = expf(vals[i] - m);
    s += vals[i];
  }
  s += __shfl_xor(s, 16, 32);
  float inv = 1.0f / s;
#pragma unroll
  for (int tile = 0; tile < 4; ++tile)
#pragma unroll
    for (int r = 0; r < 8; ++r) {
      int c = tile * 16 + r + (ty ? 8 : 0);
      red[wave][c][p] = vals[tile * 8 + r] * inv;
    }
  __syncthreads();

  // Cross-phase max == 2x2 maxpool; coalesced store of 64ch x 16px tile.
#pragma unroll
  for (int j = 0; j < 8; ++j) {
    int i  = t * 8 + j;
    int c  = i >> 4;
    int pp = i & 15;
    float v = fmaxf(fmaxf(red[0][c][pp], red[1][c][pp]),
                    fmaxf(red[2][c][pp], red[3][c][pp]));
    out[(((size_t)n * 64 + c) * 64 + y) * 64 + x0 + pp] = v;
  }
}

// ---------------------------------------------------------------------------
extern "C" void kernel_launch(void* const* d_in, const int* in_sizes, int n_in,
                              void* d_out, int out_size, void* d_ws,
                              size_t ws_size, hipStream_t stream) {
  const float* x     = (const float*)d_in[0];
  const float* w     = (const float*)d_in[1];
  const float* cb    = (const float*)d_in[2];
  const float* gmm   = (const float*)d_in[3];
  const float* bta   = (const float*)d_in[4];
  const float* rmean = (const float*)d_in[5];
  const float* rvar  = (const float*)d_in[6];
  float* w2 = (float*)d_ws;  // needs 4*64*4*32*2*4 = 256 KB of scratch

  relayout_weights<<<128, 256, 0, stream>>>(w, w2);
  fused_tconv_bn_softmax_pool<<<32 * 64 * 4, 128, 0, stream>>>(
      x, w2, cb, gmm, bta, rmean, rvar, (float*)d_out);
}